// MambaG2G_4947802325630
// MI455X (gfx1250) — compile-verified
//
#include <hip/hip_runtime.h>
#include <cstddef>

typedef __attribute__((ext_vector_type(2))) float v2f;
typedef __attribute__((ext_vector_type(8))) float v8f;

#define T_LEN 2048
#define N_B   96
#define D_M   96
#define D_I   192
#define D_S   16
#define SLD   97   // padded LDS stride for 96-wide tiles

// ---- CDNA5 async global->LDS copy (ASYNCcnt-tracked) ----------------------
#if defined(__AMDGCN__)
#define ASYNC_CP(dst, src)                                                    \
  asm volatile("global_load_async_to_lds_b32 %0, %1, off"                     \
               :: "v"((unsigned)(unsigned long long)(dst)),                   \
                  "v"((unsigned long long)(src)) : "memory")
#define ASYNC_WAIT() asm volatile("s_wait_asynccnt 0x0" ::: "memory")
#else
#define ASYNC_CP(dst, src) do { *(dst) = *(src); } while (0)
#define ASYNC_WAIT() do { } while (0)
#endif

__device__ __forceinline__ float sigmoidf_(float v) { return 1.f / (1.f + __expf(-v)); }
__device__ __forceinline__ float siluf_(float v)    { return v * sigmoidf_(v); }
__device__ __forceinline__ float softplusf_(float v){ return v > 20.f ? v : log1pf(__expf(v)); }
__device__ __forceinline__ float eluf_(float v)     { return v > 0.f ? v : (__expf(v) - 1.f); }

// One 16x16 output tile, f32 WMMA over K (multiple of 4).
// A fragment: lane l holds row m=l&15, ks (l>>4)*2 + {0,1} per step (ISA 16x4 layout).
// B fragment: lane l holds col n=l&15, same k pairing.
template <typename LA, typename LB>
__device__ __forceinline__ v8f wmma_k(int K, LA la, LB lb, v8f acc) {
  const int lane = threadIdx.x & 31;
  const int half = lane >> 4;
  const int i    = lane & 15;
  #pragma unroll 4
  for (int k = 0; k < K; k += 4) {
    const int kb = k + half * 2;
    v2f a, b;
    a.x = la(i, kb);     a.y = la(i, kb + 1);
    b.x = lb(kb, i);     b.y = lb(kb + 1, i);
    acc = __builtin_amdgcn_wmma_f32_16x16x4_f32(false, a, false, b, (short)0, acc,
                                                false, false);
  }
  return acc;
}

// ---------------------------------------------------------------------------
// Graph block: per t -> agg = mask^T*msg (WMMA), y=X+agg, h1=relu(y W1^T+b1),
// h = h1 W2^T + b2 + X, batchnorm over the 96 rows, write hbn[t].
// ---------------------------------------------------------------------------
__global__ void k_graph(const float* __restrict__ input, const float* __restrict__ edge,
                        const float* __restrict__ gW1, const float* __restrict__ gb1,
                        const float* __restrict__ gW2, const float* __restrict__ gb2,
                        const float* __restrict__ gamma, const float* __restrict__ beta,
                        float* __restrict__ hbn) {
  extern __shared__ float sm[];
  float* sX  = sm;                    // 96*97
  float* sY  = sX + N_B * SLD;        // 96*97
  float* sH  = sY + N_B * SLD;        // 96*97
  float* sE  = sH + N_B * SLD;        // 96
  float* sMu = sE + N_B;              // 96
  float* sIv = sMu + N_B;             // 96
  const int t = blockIdx.x;
  const int tid = threadIdx.x;
  const int wave = tid >> 5;
  const int lane = tid & 31, half = lane >> 4, li = lane & 15;

  for (int idx = tid; idx < N_B * D_M; idx += blockDim.x) {
    int n = idx / D_M, d = idx % D_M;
    ASYNC_CP(&sX[n * SLD + d], &input[(size_t)n * T_LEN * D_M + (size_t)t * D_M + d]);
  }
  if (tid < D_M) sE[tid] = edge[D_M + tid];   // edge_emb[1]
  ASYNC_WAIT();
  __syncthreads();

  // GEMM 1: y = X + mask^T @ relu(X + e1)
  for (int tile = wave; tile < 36; tile += 8) {
    const int c0 = (tile / 6) * 16, d0 = (tile % 6) * 16;
    v8f acc = {};
    acc = wmma_k(96,
      [&](int m, int k) { return sX[k * SLD + c0 + m] != 0.f ? 1.f : 0.f; },
      [&](int k, int n) { float v = sX[k * SLD + d0 + n] + sE[d0 + n];
                          return v > 0.f ? v : 0.f; },
      acc);
    #pragma unroll
    for (int r = 0; r < 8; ++r) {
      int row = c0 + r + 8 * half, col = d0 + li;
      sY[row * SLD + col] = acc[r] + sX[row * SLD + col];
    }
  }
  __syncthreads();

  // GEMM 2: h1 = relu(y @ W1^T + b1)
  for (int tile = wave; tile < 36; tile += 8) {
    const int r0 = (tile / 6) * 16, o0 = (tile % 6) * 16;
    v8f acc = {};
    acc = wmma_k(96,
      [&](int m, int k) { return sY[(r0 + m) * SLD + k]; },
      [&](int k, int n) { return gW1[(o0 + n) * D_M + k]; },
      acc);
    #pragma unroll
    for (int r = 0; r < 8; ++r) {
      int row = r0 + r + 8 * half, col = o0 + li;
      float v = acc[r] + gb1[col];
      sH[row * SLD + col] = v > 0.f ? v : 0.f;
    }
  }
  __syncthreads();

  // GEMM 3: h = h1 @ W2^T + b2 + X
  for (int tile = wave; tile < 36; tile += 8) {
    const int r0 = (tile / 6) * 16, o0 = (tile % 6) * 16;
    v8f acc = {};
    acc = wmma_k(96,
      [&](int m, int k) { return sH[(r0 + m) * SLD + k]; },
      [&](int k, int n) { return gW2[(o0 + n) * D_M + k]; },
      acc);
    #pragma unroll
    for (int r = 0; r < 8; ++r) {
      int row = r0 + r + 8 * half, col = o0 + li;
      sY[row * SLD + col] = acc[r] + gb2[col] + sX[row * SLD + col];
    }
  }
  __syncthreads();

  // BatchNorm over rows (axis N) per column
  if (tid < D_M) {
    float s = 0.f, ss = 0.f;
    for (int r = 0; r < N_B; ++r) { float v = sY[r * SLD + tid]; s += v; ss += v * v; }
    float mu = s * (1.f / N_B);
    sMu[tid] = mu;
    sIv[tid] = rsqrtf(ss * (1.f / N_B) - mu * mu + 1e-5f);
  }
  __syncthreads();
  for (int idx = tid; idx < N_B * D_M; idx += blockDim.x) {
    int r = idx / D_M, c = idx % D_M;
    float v = (sY[r * SLD + c] - sMu[c]) * sIv[c] * gamma[c] + beta[c];
    hbn[(size_t)t * N_B * D_M + idx] = v;
  }
}

// ---------------------------------------------------------------------------
// Generic WMMA GEMM: C[M,Nc] = act(A[M,K] @ W^T + bias), W row-major [Nc,K].
// amode: 0 = A + r*lda + k ; 1 = input[N,T,D] read as row r = t*96+n.
// epi:   0 = C[row*ldc+col] ; 1 = z[n][t][col] = v + add[row*96+col] (row=t*96+n)
// act:   0 none ; 2 elu(tanh(v)) ; 3 elu(v)+1+1e-14
// ---------------------------------------------------------------------------
__global__ void k_gemm(const float* __restrict__ A, int lda, int amode,
                       const float* __restrict__ W, int ldw,
                       const float* __restrict__ bias,
                       float* __restrict__ C, int ldc,
                       int M, int Nc, int K, int act,
                       int epi, const float* __restrict__ add) {
  __shared__ float As[64 * 33];
  __shared__ float Bs[32 * 65];
  const int tid = threadIdx.x;
  const int wave = tid >> 5, lane = tid & 31, half = lane >> 4, li = lane & 15;
  const int r0 = blockIdx.x * 64;
  const int n0 = blockIdx.y * 64;
  v8f acc[2] = {{}, {}};

  for (int kk = 0; kk < K; kk += 32) {
    __syncthreads();
    // A chunk 64x32 via async copies (OOB lanes store zero through LDS path)
    for (int idx = tid; idx < 64 * 32; idx += 256) {
      int r = idx >> 5, k = idx & 31;
      int rg = r0 + r, kg = kk + k;
      float* dst = &As[r * 33 + k];
      if (rg < M) {
        const float* src;
        if (amode == 0) src = &A[(size_t)rg * lda + kg];
        else {
          int n = rg % 96, t = rg / 96;
          src = &A[(size_t)n * (T_LEN * D_M) + (size_t)t * D_M + kg];
        }
        ASYNC_CP(dst, src);
      } else {
        *dst = 0.f;
      }
    }
    // B chunk 32x64 (transposed) via async copies
    for (int idx = tid; idx < 32 * 64; idx += 256) {
      int n = idx >> 5, k = idx & 31;
      int ng = n0 + n;
      float* dst = &Bs[k * 65 + n];
      if (ng < Nc) ASYNC_CP(dst, &W[(size_t)ng * ldw + kk + k]);
      else         *dst = 0.f;
    }
    ASYNC_WAIT();
    __syncthreads();
    #pragma unroll
    for (int j = 0; j < 2; ++j) {
      const int tt = wave * 2 + j;
      const int mt = (tt >> 2) * 16, nt = (tt & 3) * 16;
      acc[j] = wmma_k(32,
        [&](int m, int k) { return As[(mt + m) * 33 + k]; },
        [&](int k, int n) { return Bs[k * 65 + nt + n]; },
        acc[j]);
    }
  }

  for (int j = 0; j < 2; ++j) {
    const int tt = wave * 2 + j;
    const int mt = (tt >> 2) * 16, nt = (tt & 3) * 16;
    #pragma unroll
    for (int r = 0; r < 8; ++r) {
      int row = r0 + mt + r + 8 * half;
      int col = n0 + nt + li;
      if (row < M && col < Nc) {
        float v = acc[j][r];
        if (bias) v += bias[col];
        if (act == 2)      { v = tanhf(v); v = eluf_(v); }
        else if (act == 3) { v = eluf_(v) + 1.f + 1e-14f; }
        if (epi == 1) {
          int n = row % 96, t = row / 96;
          C[(size_t)n * (T_LEN * D_M) + (size_t)t * D_M + col] =
              v + add[(size_t)row * 96 + col];
        } else {
          C[(size_t)row * ldc + col] = v;
        }
      }
    }
  }
}

// Depthwise causal conv(4) + SiLU, in place on the x slice of xz [.,384].
__global__ void k_conv(float* __restrict__ xz, const float* __restrict__ convw,
                       const float* __restrict__ convb, int B, int L) {
  int gid = blockIdx.x * blockDim.x + threadIdx.x;
  if (gid >= B * D_I) return;
  int b = gid / D_I, ch = gid % D_I;
  const float w0 = convw[ch * 4 + 0], w1 = convw[ch * 4 + 1];
  const float w2 = convw[ch * 4 + 2], w3 = convw[ch * 4 + 3];
  const float bb = convb[ch];
  float x0 = 0.f, x1 = 0.f, x2 = 0.f;
  size_t base = (size_t)b * L * 384 + ch;
  for (int l = 0; l < L; ++l) {
    if (l + 8 < L) __builtin_prefetch(&xz[base + (size_t)(l + 8) * 384], 0, 1);
    float x3 = xz[base + (size_t)l * 384];
    float v = x0 * w0 + x1 * w1 + x2 * w2 + x3 * w3 + bb;
    xz[base + (size_t)l * 384] = siluf_(v);
    x0 = x1; x1 = x2; x2 = x3;
  }
}

// Selective scan: one block per sequence, thread d keeps h[16] in registers.
// dt computed in-scan from xdb[:6]; per-step xdb staged by async LDS copies
// (double-buffered, transfer overlaps the recurrence); output overwrites x.
__global__ void k_scan(float* __restrict__ xz, const float* __restrict__ xdb,
                       const float* __restrict__ Alog, const float* __restrict__ Wdt,
                       const float* __restrict__ bdt, const float* __restrict__ Dp,
                       int L) {
  __shared__ float sx[2][40];
  const int b = blockIdx.x;
  const int d = threadIdx.x;          // 0..191
  float Ad[D_S];
  #pragma unroll
  for (int n = 0; n < D_S; ++n) Ad[n] = -__expf(Alog[d * D_S + n]);
  float wd[6];
  #pragma unroll
  for (int j = 0; j < 6; ++j) wd[j] = Wdt[d * 6 + j];
  const float bd = bdt[d], Dd = Dp[d];
  float h[D_S];
  #pragma unroll
  for (int n = 0; n < D_S; ++n) h[n] = 0.f;
  const size_t rowbase = (size_t)b * L;
  if (d < 38) ASYNC_CP(&sx[0][d], &xdb[rowbase * 40 + d]);
  ASYNC_WAIT();
  __syncthreads();
  for (int t = 0; t < L; ++t) {
    const float* cx = sx[t & 1];
    if (t + 1 < L && d < 38)
      ASYNC_CP(&sx[(t + 1) & 1][d], &xdb[(rowbase + t + 1) * 40 + d]);
    float s = bd;
    #pragma unroll
    for (int j = 0; j < 6; ++j) s += cx[j] * wd[j];
    const float dtv = softplusf_(s);
    const size_t ro = (rowbase + t) * 384;
    const float xv = xz[ro + d];
    const float zv = xz[ro + D_I + d];
    const float dx = dtv * xv;
    float y = 0.f;
    #pragma unroll
    for (int n = 0; n < D_S; ++n) {
      h[n] = __expf(dtv * Ad[n]) * h[n] + dx * cx[6 + n];
      y += h[n] * cx[22 + n];
    }
    xz[ro + d] = (y + Dd * xv) * siluf_(zv);
    ASYNC_WAIT();
    __syncthreads();
  }
}

// rbar[n,k] = mean_t yfinal[n*T+t, k]  (k < 192)
__global__ void k_meanrows(const float* __restrict__ xz, float* __restrict__ rbar) {
  int gid = blockIdx.x * blockDim.x + threadIdx.x;
  if (gid >= N_B * D_I) return;
  int n = gid / D_I, k = gid % D_I;
  size_t base = (size_t)n * T_LEN * 384 + k;
  float s = 0.f;
  for (int t = 0; t < T_LEN; ++t) {
    if (t + 16 < T_LEN) __builtin_prefetch(&xz[base + (size_t)(t + 16) * 384], 0, 1);
    s += xz[base + (size_t)t * 384];
  }
  rbar[gid] = s * (1.f / T_LEN);
}

extern "C" void kernel_launch(void* const* d_in, const int* in_sizes, int n_in,
                              void* d_out, int out_size, void* d_ws, size_t ws_size,
                              hipStream_t stream) {
  (void)in_sizes; (void)n_in; (void)out_size; (void)ws_size;
  const float* input = (const float*)d_in[0];
  const float* edge  = (const float*)d_in[1];
  const float* gW1 = (const float*)d_in[2];
  const float* gb1 = (const float*)d_in[3];
  const float* gW2 = (const float*)d_in[4];
  const float* gb2 = (const float*)d_in[5];
  const float* gamma = (const float*)d_in[6];
  const float* beta  = (const float*)d_in[7];
  const float* cm[9]; for (int i = 0; i < 9; ++i) cm[i] = (const float*)d_in[8 + i];
  const float* mm[9]; for (int i = 0; i < 9; ++i) mm[i] = (const float*)d_in[17 + i];
  const float* outW = (const float*)d_in[26];
  const float* outb = (const float*)d_in[27];
  const float* muW  = (const float*)d_in[28];
  const float* mub  = (const float*)d_in[29];
  const float* sgW  = (const float*)d_in[30];
  const float* sgb  = (const float*)d_in[31];
  float* out = (float*)d_out;
  float* ws  = (float*)d_ws;

  const size_t NT = (size_t)T_LEN * N_B;          // 196608 rows for both mambas
  float* XZ   = ws;                               // NT*384
  float* XDB  = XZ + NT * 384;                    // NT*40
  float* HBN  = XDB + NT * 40;                    // T*N*D
  float* Z    = HBN + (size_t)T_LEN * N_B * D_M;  // N*T*D
  float* RBAR = Z + (size_t)T_LEN * N_B * D_M;    // 96*192
  float* EV   = RBAR + 96 * 192;                  // 96*96

  // Phase 1: graph + GNN + BN  (one block per t)
  const int smem = (3 * N_B * SLD + 3 * N_B) * (int)sizeof(float);
  (void)hipFuncSetAttribute((const void*)k_graph,
                            hipFuncAttributeMaxDynamicSharedMemorySize, smem);
  k_graph<<<T_LEN, 256, smem, stream>>>(input, edge, gW1, gb1, gW2, gb2,
                                        gamma, beta, HBN);

  // Phase 2: channel mamba (batch=T, len=96) on X = transpose(input)
  k_gemm<<<dim3(3072, 6), 256, 0, stream>>>(input, 0, 1, cm[0], 96, nullptr,
                                            XZ, 384, (int)NT, 384, 96, 0, 0, nullptr);
  k_conv<<<(T_LEN * D_I + 255) / 256, 256, 0, stream>>>(XZ, cm[1], cm[2], T_LEN, 96);
  k_gemm<<<dim3(3072, 1), 256, 0, stream>>>(XZ, 384, 0, cm[3], 192, nullptr,
                                            XDB, 40, (int)NT, 38, 192, 0, 0, nullptr);
  k_scan<<<T_LEN, 192, 0, stream>>>(XZ, XDB, cm[6], cm[4], cm[5], cm[7], 96);
  // m = yfinal @ cm_Wout^T ; z[n][t][:] = hbn[t][n][:] + m[t][n][:]
  k_gemm<<<dim3(3072, 2), 256, 0, stream>>>(XZ, 384, 0, cm[8], 192, nullptr,
                                            Z, 96, (int)NT, 96, 192, 0, 1, HBN);

  // Phase 3: time mamba (batch=N, len=T) on z
  k_gemm<<<dim3(3072, 6), 256, 0, stream>>>(Z, 96, 0, mm[0], 96, nullptr,
                                            XZ, 384, (int)NT, 384, 96, 0, 0, nullptr);
  k_conv<<<(N_B * D_I + 255) / 256, 256, 0, stream>>>(XZ, mm[1], mm[2], N_B, T_LEN);
  k_gemm<<<dim3(3072, 1), 256, 0, stream>>>(XZ, 384, 0, mm[3], 192, nullptr,
                                            XDB, 40, (int)NT, 38, 192, 0, 0, nullptr);
  k_scan<<<N_B, 192, 0, stream>>>(XZ, XDB, mm[6], mm[4], mm[5], mm[7], T_LEN);

  // Phase 4: e = (mean_t yfinal) @ m_Wout^T  (linearity of mean + GEMM)
  k_meanrows<<<(N_B * D_I + 255) / 256, 256, 0, stream>>>(XZ, RBAR);
  k_gemm<<<dim3(2, 2), 256, 0, stream>>>(RBAR, 192, 0, mm[8], 192, nullptr,
                                         EV, 96, 96, 96, 192, 0, 0, nullptr);

  // Head: x = elu(tanh(e@outW^T+b)) -> d_out[0:9216]; mu, sigma follow.
  k_gemm<<<dim3(2, 2), 256, 0, stream>>>(EV, 96, 0, outW, 96, outb,
                                         out, 96, 96, 96, 96, 2, 0, nullptr);
  k_gemm<<<dim3(2, 1), 256, 0, stream>>>(out, 96, 0, muW, 96, mub,
                                         out + 9216, 64, 96, 64, 96, 0, 0, nullptr);
  k_gemm<<<dim3(2, 1), 256, 0, stream>>>(out, 96, 0, sgW, 96, sgb,
                                         out + 9216 + 6144, 64, 96, 64, 96, 3, 0, nullptr);
}